// HCRLayer_14577119003096
// MI455X (gfx1250) — compile-verified
//
#include <hip/hip_runtime.h>
#include <hip/hip_bf16.h>

typedef float v2f __attribute__((ext_vector_type(2)));
typedef float v8f __attribute__((ext_vector_type(8)));

#define DEG   16     // M+1 basis functions per dim
#define KDIM  4096   // (M+1)^3 features
#define NOUT  1024
#define BM    128
#define BN    128
#define BK    32
#define PROW  144    // paired-row stride: 2*144 == 32 (mod 64) -> lane-halves on disjoint banks

__launch_bounds__(256)
__global__ void hcr_legendre_gemm(const float* __restrict__ X,
                                  const float* __restrict__ C,
                                  float* __restrict__ Out)
{
    __shared__ float basis_s[3][DEG][BM];     // 24.0 KB
    __shared__ v2f   A2[BK / 2][PROW];        // 18.0 KB : A2[q][m] = {A[2q][m], A[2q+1][m]}
    __shared__ v2f   B2[BK / 2][PROW];        // 18.0 KB : B2[q][n] = {B[2q][n], B[2q+1][n]}
    __shared__ float corr_s[BN];              //  0.5 KB

    const int t    = threadIdx.x;
    const int lane = t & 31;
    const int wave = t >> 5;
    const int m0   = blockIdx.y * BM;         // batch tile origin
    const int n0   = blockIdx.x * BN;         // output tile origin

    // ---------------- per-block basis: f_k(x) = sqrt(2k+1) * P_k(2x-1) ----------------
    for (int task = t; task < 3 * BM; task += 256) {
        const int d = task >> 7;              // dim 0..2 (wave-uniform)
        const int r = task & (BM - 1);
        const float x  = X[(m0 + r) * 3 + d];
        const float tt = 2.0f * x - 1.0f;
        float pkm1 = 1.0f;
        float pk   = tt;
        basis_s[d][0][r] = 1.0f;
        basis_s[d][1][r] = tt * 1.7320508075688772f;   // sqrt(3)
#pragma unroll
        for (int k = 1; k < DEG - 1; ++k) {
            const float pkp1 = ((2.0f * k + 1.0f) * tt * pk - (float)k * pkm1)
                               / (float)(k + 1);
            pkm1 = pk;
            pk   = pkp1;
            basis_s[d][k + 1][r] = pk * sqrtf(2.0f * (float)(k + 1) + 1.0f); // const-folded
        }
    }
    // density = 1 + Bfull@C^T - Bfull[:,0]*C[:,0], and Bfull[:,0]==1 identically
    for (int n = t; n < BN; n += 256)
        corr_s[n] = 1.0f - C[(size_t)(n0 + n) * KDIM];
    __syncthreads();

    // ---------------- wave tiling: 4 (M) x 2 (N) waves; each wave 32x64 = 2x4 frags ---
    const int waveM = wave & 3;
    const int waveN = wave >> 2;
    const int lm    = lane & 15;
    const int h     = lane >> 4;              // lane half: pair q vs q+1

    v8f acc[2][4];
#pragma unroll
    for (int mt = 0; mt < 2; ++mt)
#pragma unroll
        for (int nt = 0; nt < 4; ++nt)
            acc[mt][nt] = (v8f){0.f, 0.f, 0.f, 0.f, 0.f, 0.f, 0.f, 0.f};

    // Async B-tile mapping: j = k-pair index (lane-fast -> 128B coalesced global runs)
    const int jB = t & 15;                    // k-pair 0..15
    const int nB = t >> 4;                    // first of 8 output rows, step 16

    // ---------------- K loop over the 4096 tensor-product features --------------------
    for (int k0 = 0; k0 < KDIM; k0 += BK) {
        // (1) kick off async global->LDS copy of the C^T tile (ASYNCcnt-tracked)
        {
            const float* Cstage = C + k0;     // SGPR64 base for GVS addressing
#pragma unroll
            for (int i = 0; i < 8; ++i) {
                const int n = nB + i * 16;
                const unsigned voff = (unsigned)(((n0 + n) * KDIM + 2 * jB)
                                                 * (int)sizeof(float));
                const unsigned lds  = (unsigned)(uintptr_t)&B2[jB][n];
                asm volatile("global_load_async_to_lds_b64 %0, %1, %2"
                             :: "v"(lds), "v"(voff), "s"(Cstage)
                             : "memory");
            }
        }

        // (2) overlap: generate paired A tile  Bfull[m, kg] = b0[i0]*b1[i1]*b2[i2]
        {
            const int m   = t & (BM - 1);
            const int kk0 = (t >> 7) * (BK / 2);   // 0 or 16 (wave-uniform)
#pragma unroll
            for (int p = 0; p < BK / 4; ++p) {
                const int kg = k0 + kk0 + 2 * p;
                const float a0 = basis_s[0][(kg >> 8) & 15][m]
                               * basis_s[1][(kg >> 4) & 15][m]
                               * basis_s[2][kg & 15][m];
                const int kg1 = kg + 1;
                const float a1 = basis_s[0][(kg1 >> 8) & 15][m]
                               * basis_s[1][(kg1 >> 4) & 15][m]
                               * basis_s[2][kg1 & 15][m];
                A2[(kk0 >> 1) + p][m] = (v2f){a0, a1};
            }
        }

        // (3) drain async transfers, then block-wide barrier
        asm volatile("s_wait_asynccnt 0" ::: "memory");
        __syncthreads();

        // (4) WMMA burst: 8 K-steps x 8 fragments
#pragma unroll
        for (int kk = 0; kk < BK; kk += 4) {
            const int q = (kk >> 1) + h;      // pair row for this lane half
            v2f afrag[2], bfrag[4];
#pragma unroll
            for (int mt = 0; mt < 2; ++mt)
                afrag[mt] = A2[q][waveM * 32 + mt * 16 + lm];   // ds_load_b64
#pragma unroll
            for (int nt = 0; nt < 4; ++nt)
                bfrag[nt] = B2[q][waveN * 64 + nt * 16 + lm];   // ds_load_b64
#pragma unroll
            for (int mt = 0; mt < 2; ++mt)
#pragma unroll
                for (int nt = 0; nt < 4; ++nt)
                    acc[mt][nt] = __builtin_amdgcn_wmma_f32_16x16x4_f32(
                        false, afrag[mt], false, bfrag[nt],
                        (short)0, acc[mt][nt], false, false);
        }
        __syncthreads();
    }

    // ---------------- epilogue: out = acc + (1 - C[n,0]) ------------------------------
#pragma unroll
    for (int mt = 0; mt < 2; ++mt) {
#pragma unroll
        for (int nt = 0; nt < 4; ++nt) {
            const int nl   = waveN * 64 + nt * 16 + lm;
            const int n    = n0 + nl;
            const float cr = corr_s[nl];
#pragma unroll
            for (int v = 0; v < 8; ++v) {
                const int m = m0 + waveM * 32 + mt * 16 + v + 8 * h;  // C/D layout
                Out[(size_t)m * NOUT + n] = acc[mt][nt][v] + cr;
            }
        }
    }
}

extern "C" void kernel_launch(void* const* d_in, const int* in_sizes, int n_in,
                              void* d_out, int out_size, void* d_ws, size_t ws_size,
                              hipStream_t stream) {
    const float* x     = (const float*)d_in[0];   // [BATCH, 3] fp32
    const float* coeff = (const float*)d_in[1];   // [1024, 4096] fp32
    float* out         = (float*)d_out;           // [BATCH, 1024] fp32

    const int batch = in_sizes[0] / 3;            // 8192
    dim3 grid(NOUT / BN, batch / BM);             // (8, 64)
    hcr_legendre_gemm<<<grid, 256, 0, stream>>>(x, coeff, out);
}